// ConvNet_5858335392257
// MI455X (gfx1250) — compile-verified
//
#include <hip/hip_runtime.h>
#include <cstdint>

// MI455X / gfx1250, wave32. FP32 WMMA (V_WMMA_F32_16X16X4_F32) for all dense
// node GEMMs; edge gather/scatter via float4-per-lane + L2-resident atomics.
// K-loops kept rolled with all independent output tiles as parallel WMMA
// accumulator chains -> high ILP at < 256 VGPRs (no s_set_vgpr_msb churn).

typedef float v2f __attribute__((ext_vector_type(2)));
typedef float v8f __attribute__((ext_vector_type(8)));

#define Nn 50000
#define Ee 800000
#define NT ((Nn + 15) / 16)  // 3125 exact

__device__ __forceinline__ v8f wmma4(v2f a, v2f b, v8f c) {
  // D(16x16) = A(16x4) * B(4x16) + C ; fp32 matrix pipe
  return __builtin_amdgcn_wmma_f32_16x16x4_f32(false, a, false, b, (short)0, c,
                                               false, false);
}

__device__ __forceinline__ float sspf(float x) {  // softplus(x) - ln(2), stable
  return fmaxf(x, 0.0f) + log1pf(__expf(-fabsf(x))) - 0.69314718056f;
}

// ---------------------------------------------------------------------------
// Kernel 1: per 16-node tile (one wave each):
//   feat[n][c] = {s_in@W0, (v_in@W1)_xyz}   (layout (N,C,4) for float4 gather)
//   sc[n]      = {(s_in (x) attr)@Wsc_s [64] , (v_in (x) attr)@Wsc_v [f*3+i]}
//   agg[n]     = 0
// ---------------------------------------------------------------------------
__global__ __launch_bounds__(128) void node_pre_kernel(
    const float* __restrict__ x, const float* __restrict__ attrs,
    const float* __restrict__ W0l, const float* __restrict__ W1l,
    const float* __restrict__ Wscs, const float* __restrict__ Wscv,
    float* __restrict__ featbuf, float* __restrict__ scbuf,
    float* __restrict__ aggbuf) {
  __shared__ float lds_x[4][16][128];
  __shared__ float lds_at[4][16][4];
  const int w = threadIdx.x >> 5;
  const int lane = threadIdx.x & 31;
  const int tl = lane & 15;
  const int th = lane >> 4;
  const int tile = blockIdx.x * 4 + w;
  const bool active = tile < NT;
  const int base = tile * 16;

  if (active) {
    const float4* xg = (const float4*)x;
    float4* ag4 = (float4*)aggbuf;
    float4 z4;
    z4.x = z4.y = z4.z = z4.w = 0.0f;
#pragma unroll
    for (int t = 0; t < 16; ++t) {
      int idx = t * 32 + lane;
      int row = idx >> 5, c4 = idx & 31;
      ((float4*)lds_x[w][row])[c4] = xg[(size_t)(base + row) * 32 + c4];
      ag4[(size_t)(base + row) * 32 + c4] = z4;  // zero aggregation buffer
    }
    if (lane < 16)
      ((float4*)lds_at[w][lane])[0] = ((const float4*)attrs)[base + lane];
  }
  __syncthreads();
  if (!active) return;

  const float* Xrow = lds_x[w][tl];
  const float at0 = lds_at[w][tl][2 * th];
  const float at1 = lds_at[w][tl][2 * th + 1];

  // ================= Phase 1: s = s_in@W0 ; v_i = v_in_i@W1 ================
  // 8 parallel accumulator chains over one rolled K=32 loop.
  {
    v8f sacc[2] = {};
    v8f vacc[3][2] = {};
#pragma unroll 1
    for (int k = 0; k < 32; k += 4) {
      const int c0 = k + 2 * th;
      v2f as;
      as.x = Xrow[c0];
      as.y = Xrow[c0 + 1];
      v2f av[3];
#pragma unroll
      for (int i = 0; i < 3; ++i) {
        av[i].x = Xrow[32 + c0 * 3 + i];
        av[i].y = Xrow[32 + (c0 + 1) * 3 + i];
      }
#pragma unroll
      for (int ft = 0; ft < 2; ++ft) {
        const float* B0 = W0l + c0 * 32 + ft * 16 + tl;
        v2f b0;
        b0.x = B0[0];
        b0.y = B0[32];
        sacc[ft] = wmma4(as, b0, sacc[ft]);
        const float* B1 = W1l + c0 * 32 + ft * 16 + tl;
        v2f b1;
        b1.x = B1[0];
        b1.y = B1[32];
#pragma unroll
        for (int i = 0; i < 3; ++i) vacc[i][ft] = wmma4(av[i], b1, vacc[i][ft]);
      }
    }
    // (node,f) mapping identical across the 4 GEMMs -> single float4 store
    float4* fb4 = (float4*)featbuf;
#pragma unroll
    for (int ft = 0; ft < 2; ++ft) {
      const int f = ft * 16 + tl;
#pragma unroll
      for (int r = 0; r < 8; ++r) {
        float4 o;
        o.x = sacc[ft][r];
        o.y = vacc[0][ft][r];
        o.z = vacc[1][ft][r];
        o.w = vacc[2][ft][r];
        fb4[(size_t)(base + r + 8 * th) * 32 + f] = o;
      }
    }
  }

  // ====== Phase 2: self-connection GEMMs, K=128 Kronecker operand ==========
  // sc_s: 4 col-tiles ; sc_v: 3 x 2 col-tiles -> 10 parallel chains.
  {
    v8f accs[4] = {};
    v8f accv[3][2] = {};
#pragma unroll 1
    for (int k = 0; k < 128; k += 4) {
      const int c = k >> 2;  // flat col = k+2th -> channel k>>2, z = 2th(+1)
      const float sv = Xrow[c];
      v2f a_s;
      a_s.x = sv * at0;
      a_s.y = sv * at1;
      v2f a_v[3];
#pragma unroll
      for (int i = 0; i < 3; ++i) {
        const float vv = Xrow[32 + c * 3 + i];
        a_v[i].x = vv * at0;
        a_v[i].y = vv * at1;
      }
      const float* Bs = Wscs + (k + 2 * th) * 64 + tl;
#pragma unroll
      for (int ft = 0; ft < 4; ++ft) {
        v2f b;
        b.x = Bs[ft * 16];
        b.y = Bs[64 + ft * 16];
        accs[ft] = wmma4(a_s, b, accs[ft]);
      }
      const float* Bv = Wscv + (k + 2 * th) * 32 + tl;
#pragma unroll
      for (int ft = 0; ft < 2; ++ft) {
        v2f b;
        b.x = Bv[ft * 16];
        b.y = Bv[32 + ft * 16];
#pragma unroll
        for (int i = 0; i < 3; ++i) accv[i][ft] = wmma4(a_v[i], b, accv[i][ft]);
      }
    }
#pragma unroll
    for (int ft = 0; ft < 4; ++ft) {
      const int f = ft * 16 + tl;
#pragma unroll
      for (int r = 0; r < 8; ++r)
        scbuf[(size_t)(base + r + 8 * th) * 160 + f] = accs[ft][r];
    }
#pragma unroll
    for (int ft = 0; ft < 2; ++ft) {
      const int f = ft * 16 + tl;
#pragma unroll
      for (int i = 0; i < 3; ++i)
#pragma unroll
        for (int r = 0; r < 8; ++r)
          scbuf[(size_t)(base + r + 8 * th) * 160 + 64 + f * 3 + i] =
              accv[i][ft][r];
    }
  }
}

// ---------------------------------------------------------------------------
// Kernel 2: one wave per edge, lane = channel c (0..31).
//   w = silu(ee@Wr1)@Wr2 ; messages; atomic scatter-add (float4 per lane).
//   1/DEG folded into the message. agg buffer (25.6MB) stays L2-resident.
// ---------------------------------------------------------------------------
__global__ __launch_bounds__(256) void edge_kernel(
    const int* __restrict__ ei, const float* __restrict__ ee,
    const float* __restrict__ ea, const float* __restrict__ Wr1l,
    const float* __restrict__ Wr2l, const float* __restrict__ featbuf,
    float* __restrict__ aggbuf) {
  __shared__ float sWr1[64];
  __shared__ float sWr2[1280];
  for (int t = threadIdx.x; t < 64; t += 256) sWr1[t] = Wr1l[t];
  for (int t = threadIdx.x; t < 1280; t += 256) sWr2[t] = Wr2l[t];
  __syncthreads();

  const int lane = threadIdx.x & 31;
  const int c = lane;
  const int j8 = lane & 7;
  int e = blockIdx.x * 8 + (threadIdx.x >> 5);
  const int nw = gridDim.x * 8;

  for (; e < Ee; e += nw) {
    const int src = ei[e];
    const int dst = ei[Ee + e];
    const float y0 = ea[(size_t)e * 4 + 0];
    const float yv0 = ea[(size_t)e * 4 + 1];
    const float yv1 = ea[(size_t)e * 4 + 2];
    const float yv2 = ea[(size_t)e * 4 + 3];

    // hidden h[j] = silu(ee . Wr1[:,j]) ; all lanes compute for j = lane&7
    const float4 eA = ((const float4*)ee)[(size_t)e * 2 + 0];
    const float4 eB = ((const float4*)ee)[(size_t)e * 2 + 1];
    float t = eA.x * sWr1[0 * 8 + j8] + eA.y * sWr1[1 * 8 + j8] +
              eA.z * sWr1[2 * 8 + j8] + eA.w * sWr1[3 * 8 + j8] +
              eB.x * sWr1[4 * 8 + j8] + eB.y * sWr1[5 * 8 + j8] +
              eB.z * sWr1[6 * 8 + j8] + eB.w * sWr1[7 * 8 + j8];
    const float h = t / (1.0f + __expf(-t));

    float w1 = 0.f, w2 = 0.f, w3 = 0.f, w4 = 0.f, w5 = 0.f;
#pragma unroll
    for (int j = 0; j < 8; ++j) {
      const float hj = __shfl(h, j, 32);
      const float* r2 = sWr2 + j * 160;
      w1 += hj * r2[c];
      w2 += hj * r2[32 + c];
      w3 += hj * r2[64 + c];
      w4 += hj * r2[96 + c];
      w5 += hj * r2[128 + c];
    }

    // gather transformed features of src: {s, vx, vy, vz} for this channel
    const float4 f = ((const float4*)featbuf)[(size_t)src * 32 + c];
    const float se = f.x, v0 = f.y, v1 = f.z, v2 = f.w;
    const float dotv = v0 * yv0 + v1 * yv1 + v2 * yv2;
    const float cx = v1 * yv2 - v2 * yv1;
    const float cy = v2 * yv0 - v0 * yv2;
    const float cz = v0 * yv1 - v1 * yv0;
    const float ms = (w1 * se * y0 + w4 * dotv) * 0.0625f;
    const float mv0 = (w2 * se * yv0 + w3 * v0 * y0 + w5 * cx) * 0.0625f;
    const float mv1 = (w2 * se * yv1 + w3 * v1 * y0 + w5 * cy) * 0.0625f;
    const float mv2 = (w2 * se * yv2 + w3 * v2 * y0 + w5 * cz) * 0.0625f;

    float* ag = aggbuf + (size_t)dst * 128 + c * 4;
    atomicAdd(ag + 0, ms);
    atomicAdd(ag + 1, mv0);
    atomicAdd(ag + 2, mv1);
    atomicAdd(ag + 3, mv2);
  }
}

// ---------------------------------------------------------------------------
// Kernel 3: out_s = agg_s@Ws + sc_s ; out_v = agg_v@Wv + sc_v ;
//   feat = ssp(out_s[:,:32]); gates = ssp(out_s[:,32:]); x += [feat, out_v*g]
// ---------------------------------------------------------------------------
__global__ __launch_bounds__(128) void node_post_kernel(
    const float* __restrict__ aggbuf, const float* __restrict__ scbuf,
    const float* __restrict__ Wsl, const float* __restrict__ Wvl,
    float* __restrict__ x) {
  __shared__ float lds_a[4][16][128];
  __shared__ float lds_g[4][16][32];
  const int w = threadIdx.x >> 5;
  const int lane = threadIdx.x & 31;
  const int tl = lane & 15;
  const int th = lane >> 4;
  const int tile = blockIdx.x * 4 + w;
  const bool active = tile < NT;
  const int base = tile * 16;

  if (active) {
    const float4* ag = (const float4*)aggbuf;
#pragma unroll
    for (int t = 0; t < 16; ++t) {
      int idx = t * 32 + lane;
      int row = idx >> 5, c4 = idx & 31;
      ((float4*)lds_a[w][row])[c4] = ag[(size_t)(base + row) * 32 + c4];
    }
  }
  __syncthreads();

  const float* Arow = lds_a[w][tl];
  if (active) {
    // out_s (16x32 * 32x64): 4 parallel chains, rolled K loop
    v8f acc[4] = {};
#pragma unroll 1
    for (int k = 0; k < 32; k += 4) {
      const int c0 = k + 2 * th;
      v2f a;
      a.x = Arow[c0 * 4];
      a.y = Arow[(c0 + 1) * 4];
      const float* Bp = Wsl + c0 * 64 + tl;
#pragma unroll
      for (int ft = 0; ft < 4; ++ft) {
        v2f b;
        b.x = Bp[ft * 16];
        b.y = Bp[64 + ft * 16];
        acc[ft] = wmma4(a, b, acc[ft]);
      }
    }
#pragma unroll
    for (int ft = 0; ft < 4; ++ft) {
      const int col = ft * 16 + tl;
#pragma unroll
      for (int r = 0; r < 8; ++r) {
        const int m = r + 8 * th;
        const int node = base + m;
        const float val = sspf(acc[ft][r] + scbuf[(size_t)node * 160 + col]);
        if (col < 32)
          x[(size_t)node * 128 + col] += val;  // scalar residual
        else
          lds_g[w][m][col - 32] = val;  // gate
      }
    }
  }
  __syncthreads();
  if (!active) return;

  // out_v_i (16x32 * 32x32): 6 parallel chains, rolled K loop
  {
    v8f acc[3][2] = {};
#pragma unroll 1
    for (int k = 0; k < 32; k += 4) {
      const int c0 = k + 2 * th;
      v2f a[3];
#pragma unroll
      for (int i = 0; i < 3; ++i) {
        a[i].x = Arow[c0 * 4 + 1 + i];
        a[i].y = Arow[(c0 + 1) * 4 + 1 + i];
      }
      const float* Bp = Wvl + c0 * 32 + tl;
#pragma unroll
      for (int ft = 0; ft < 2; ++ft) {
        v2f b;
        b.x = Bp[ft * 16];
        b.y = Bp[32 + ft * 16];
#pragma unroll
        for (int i = 0; i < 3; ++i) acc[i][ft] = wmma4(a[i], b, acc[i][ft]);
      }
    }
#pragma unroll
    for (int ft = 0; ft < 2; ++ft) {
      const int f = ft * 16 + tl;
#pragma unroll
      for (int i = 0; i < 3; ++i) {
#pragma unroll
        for (int r = 0; r < 8; ++r) {
          const int m = r + 8 * th;
          const int node = base + m;
          const float val =
              acc[i][ft][r] + scbuf[(size_t)node * 160 + 64 + f * 3 + i];
          x[(size_t)node * 128 + 32 + f * 3 + i] += val * lds_g[w][m][f];
        }
      }
    }
  }
}

// ---------------------------------------------------------------------------
extern "C" void kernel_launch(void* const* d_in, const int* in_sizes, int n_in,
                              void* d_out, int out_size, void* d_ws,
                              size_t ws_size, hipStream_t stream) {
  (void)in_sizes; (void)n_in; (void)out_size; (void)ws_size;
  const float* node_features = (const float*)d_in[0];
  const float* node_attrs    = (const float*)d_in[1];
  const int*   edge_index    = (const int*)d_in[2];
  const float* edge_emb      = (const float*)d_in[3];
  const float* edge_attrs    = (const float*)d_in[4];
  const float* W0    = (const float*)d_in[5];
  const float* W1    = (const float*)d_in[6];
  const float* Wr1   = (const float*)d_in[7];
  const float* Wr2   = (const float*)d_in[8];
  const float* Ws    = (const float*)d_in[9];
  const float* Wv    = (const float*)d_in[10];
  const float* Wsc_s = (const float*)d_in[11];
  const float* Wsc_v = (const float*)d_in[12];

  float* x = (float*)d_out;                       // evolving node state (N,128)
  float* ws = (float*)d_ws;
  float* featbuf = ws;                            // (N, C, 4) {s,vx,vy,vz}
  float* aggbuf  = featbuf + (size_t)Nn * 128;    // (N, C, 4) atomic target
  float* scbuf   = aggbuf  + (size_t)Nn * 128;    // (N, 160) {sc_s, sc_v}

  hipMemcpyAsync(x, node_features, (size_t)Nn * 128 * sizeof(float),
                 hipMemcpyDeviceToDevice, stream);

  const int nodeBlocks = (NT + 3) / 4;  // 4 waves (tiles) per 128-thread block
  for (int l = 0; l < 2; ++l) {
    node_pre_kernel<<<nodeBlocks, 128, 0, stream>>>(
        x, node_attrs, W0 + l * 1024, W1 + l * 1024, Wsc_s + l * 8192,
        Wsc_v + l * 4096, featbuf, scbuf, aggbuf);
    edge_kernel<<<4096, 256, 0, stream>>>(edge_index, edge_emb, edge_attrs,
                                          Wr1 + l * 64, Wr2 + l * 1280,
                                          featbuf, aggbuf);
    node_post_kernel<<<nodeBlocks, 128, 0, stream>>>(
        aggbuf, scbuf, Ws + l * 2048, Wv + l * 1024, x);
  }
}